// InteractionLayer_30631706755974
// MI455X (gfx1250) — compile-verified
//
#include <hip/hip_runtime.h>

// Problem constants (from the reference)
#define NFEAT  10000
#define NF4    2500      // NFEAT / 4
#define NBATCH 4096
#define LAT    64

// Workspace layout in floats
#define WS_S   0         // s[k]  = sum_b X[b,k]          (10000)
#define WS_S2  10000     // s2[k] = sum_b X[b,k]^2        (10000)
#define WS_T   20000     // t[f]  = sum_k s[k]*V[k,f]     (64)
#define WS_Q   20064     // q     = sum_k s2[k]*v2[k]     (1)
#define WS_N   20065

typedef float v2f __attribute__((ext_vector_type(2)));
typedef float v4f __attribute__((ext_vector_type(4)));
typedef float v8f __attribute__((ext_vector_type(8)));

// ---------------------------------------------------------------- zero ws
__global__ void zero_ws_kernel(float* __restrict__ ws) {
  int i = blockIdx.x * blockDim.x + threadIdx.x;
  if (i < WS_N) ws[i] = 0.0f;
}

// ------------------------------------------------- bulk pass over X (BW-bound)
// grid = (10, 64): x tiles columns (256 float4 lanes/block), y tiles 64-row strips.
// Nontemporal B128 loads: X is single-use, don't pollute caches.
__global__ __launch_bounds__(256) void colreduce_kernel(const float* __restrict__ X,
                                                        float* __restrict__ s,
                                                        float* __restrict__ s2) {
  const int k4 = blockIdx.x * 256 + threadIdx.x;   // float4 column index
  if (k4 >= NF4) return;
  const int rows = NBATCH / 64;                    // 64 rows per y-block
  const int r0 = blockIdx.y * rows;
  const v4f* __restrict__ X4 = (const v4f*)X;      // row stride 40000B -> 16B aligned

  float sx = 0.f, sy = 0.f, sz = 0.f, sw = 0.f;
  float qx = 0.f, qy = 0.f, qz = 0.f, qw = 0.f;
#pragma unroll 4
  for (int r = 0; r < rows; ++r) {
    v4f v = __builtin_nontemporal_load(&X4[(size_t)(r0 + r) * NF4 + k4]);
    sx += v.x; sy += v.y; sz += v.z; sw += v.w;
    qx += v.x * v.x; qy += v.y * v.y; qz += v.z * v.z; qw += v.w * v.w;
  }
  const int k = 4 * k4;
  atomicAdd(&s[k + 0], sx);  atomicAdd(&s[k + 1], sy);
  atomicAdd(&s[k + 2], sz);  atomicAdd(&s[k + 3], sw);
  atomicAdd(&s2[k + 0], qx); atomicAdd(&s2[k + 1], qy);
  atomicAdd(&s2[k + 2], qz); atomicAdd(&s2[k + 3], qw);
}

// ------------------------------------------------- q = sum_k s2[k] * ||V[k,:]||^2
__global__ __launch_bounds__(256) void q_kernel(const float* __restrict__ V,
                                                const float* __restrict__ s2,
                                                float* __restrict__ q) {
  __shared__ float red[256];
  const int k = blockIdx.x * 256 + threadIdx.x;
  float part = 0.f;
  if (k < NFEAT) {
    const v4f* __restrict__ Vr = (const v4f*)(V + (size_t)k * LAT);  // 256B row stride
    float v2 = 0.f;
#pragma unroll
    for (int i = 0; i < LAT / 4; ++i) {
      v4f v = Vr[i];
      v2 += v.x * v.x + v.y * v.y + v.z * v.z + v.w * v.w;
    }
    part = s2[k] * v2;
  }
  red[threadIdx.x] = part;
  __syncthreads();
  for (int off = 128; off > 0; off >>= 1) {
    if (threadIdx.x < off) red[threadIdx.x] += red[threadIdx.x + off];
    __syncthreads();
  }
  if (threadIdx.x == 0) atomicAdd(q, red[0]);
}

// ------------------------------------------------- t[f] = sum_k s[k]*V[k,f] via WMMA
// A[m,kk] = s[k0+kk] * V[k0+kk, f0+m]  (documented 16x4 f32 A layout)
// B       = all-ones (layout-invariant) -> D[m,n] += sum_kk A[m,kk]
// 512 waves: wave w -> f-tile (w&3), K-strip (w>>2) of 20 steps (K=4 each).
__global__ __launch_bounds__(256) void t_wmma_kernel(const float* __restrict__ V,
                                                     const float* __restrict__ s,
                                                     float* __restrict__ t) {
  const int lane = threadIdx.x & 31;
  const int wave = (blockIdx.x * 256 + threadIdx.x) >> 5;  // 0..511
  const int f0 = (wave & 3) * 16;
  const int ks = wave >> 2;                                // 0..127
  const int m = lane & 15;
  const int sel = lane >> 4;                               // 0: K={0,1}, 1: K={2,3}

  v8f acc = {};
  v2f ones = {1.0f, 1.0f};

  for (int i = 0; i < 20; ++i) {          // wave-uniform trip count -> EXEC stays all-1s
    const int j = ks * 20 + i;            // K-step index, 4 k's per step
    if (j >= NF4) break;
    const int ka = 4 * j + sel * 2;
    v2f a;
    a.x = s[ka]     * V[(size_t)ka * LAT + f0 + m];
    a.y = s[ka + 1] * V[(size_t)(ka + 1) * LAT + f0 + m];
    // D = A x B + C, full f32: v_wmma_f32_16x16x4_f32
    acc = __builtin_amdgcn_wmma_f32_16x16x4_f32(false, a, false, ones,
                                                (short)0, acc, false, false);
  }

  // D layout: VGPR r -> M=r (lanes 0-15) / M=8+r (lanes 16-31); all N columns equal.
  if ((lane & 15) == 0) {
    const int fb = f0 + ((lane >> 4) << 3);
#pragma unroll
    for (int r = 0; r < 8; ++r) atomicAdd(&t[fb + r], acc[r]);
  }
}

// ------------------------------------------------- scalar finalize
__global__ void finalize_kernel(const float* __restrict__ t,
                                const float* __restrict__ q,
                                float* __restrict__ out) {
  if (threadIdx.x == 0) {
    float st = 0.f;
    for (int f = 0; f < LAT; ++f) st += t[f] * t[f];
    out[0] = 0.5f * (st - q[0]);
  }
}

extern "C" void kernel_launch(void* const* d_in, const int* in_sizes, int n_in,
                              void* d_out, int out_size, void* d_ws, size_t ws_size,
                              hipStream_t stream) {
  (void)in_sizes; (void)n_in; (void)out_size; (void)ws_size;
  const float* X = (const float*)d_in[0];   // [4096, 10000] f32
  const float* V = (const float*)d_in[1];   // [10000, 64]  f32
  float* out = (float*)d_out;               // scalar f32
  float* ws = (float*)d_ws;
  float* s  = ws + WS_S;
  float* s2 = ws + WS_S2;
  float* t  = ws + WS_T;
  float* q  = ws + WS_Q;

  zero_ws_kernel<<<(WS_N + 255) / 256, 256, 0, stream>>>(ws);
  colreduce_kernel<<<dim3((NF4 + 255) / 256, 64), 256, 0, stream>>>(X, s, s2);
  q_kernel<<<(NFEAT + 255) / 256, 256, 0, stream>>>(V, s2, q);
  t_wmma_kernel<<<64, 256, 0, stream>>>(V, s, t);
  finalize_kernel<<<1, 32, 0, stream>>>(t, q, out);
}